// RelationalGATEncoder_70076686401549
// MI455X (gfx1250) — compile-verified
//
#include <hip/hip_runtime.h>
#include <hip/hip_bf16.h>

// ---------------------------------------------------------------------------
// RelationalGATEncoder for MI455X (gfx1250, wave32).
// Dense projections via V_WMMA_F32_16X16X4_F32 with LDS-staged B fragments;
// edge softmax/aggregation via L2-resident atomic passes (xp = 102MB < 192MB L2).
// ---------------------------------------------------------------------------

#define NN   50000   // num entities
#define RR   8       // relations
#define EE   200000  // edges per relation
#define DD   128     // emb dim
#define HID_ 128
#define HH   4       // heads layer 1
#define OUTD 256
#define F1   (HH * HID_)   // 512
#define KCHUNK 128         // K tile staged in LDS (32KB)

typedef __attribute__((ext_vector_type(2))) float v2f;
typedef __attribute__((ext_vector_type(8))) float v8f;

#define WMMA_F32(a, b, c) \
    __builtin_amdgcn_wmma_f32_16x16x4_f32(false, (a), false, (b), (short)0, (c), false, false)

// ---------------------------------------------------------------------------
// WMMA GEMM: C[M,Nn] = A[M,K] @ B[K,Nn]  (row major, fp32), optional +=.
// Block = 8 waves; block covers 16 M-tiles x 64 N-cols.
// Each wave: 2 M-tiles (16x64 each), B strip staged in LDS in fragment order:
//   ldsB[((kstep*32 + lane)*4 + sub)] = { B[k+2*half][n], B[k+2*half+1][n] }
// Inner loop is branch-free: tile indices are clamped for loads (in-bounds),
// validity checked only at the store. Per K-step: 2 x ds_load_b128 (B)
// + 2 x global b64 (A) -> 8 WMMAs. Requires M%16==0, Nn%64==0, K%128==0.
// ---------------------------------------------------------------------------
__global__ __launch_bounds__(256) void wmma_gemm_f32_kernel(
    const float* __restrict__ A, const float* __restrict__ B,
    float* __restrict__ C, int M, int Nn, int K, int accumulate)
{
    __shared__ float2 ldsB[(KCHUNK / 4) * 32 * 4];   // 32 KB

    const int tid  = threadIdx.x;
    const int lane = tid & 31;
    const int wv   = tid >> 5;
    const int half = lane >> 4;
    const int l15  = lane & 15;
    const int nbase = blockIdx.x * 64;
    const int mtiles = M >> 4;
    const int mt0 = blockIdx.y * 16 + wv * 2;
    const int mt1 = mt0 + 1;
    // clamp for loads (always in-bounds); validity used only for stores
    const int mtLast = mtiles - 1;
    const int m0 = mt0 < mtLast ? mt0 : mtLast;
    const int m1 = mt1 < mtLast ? mt1 : mtLast;

    const float* Ap0 = A + (size_t)(m0 * 16 + l15) * K;
    const float* Ap1 = A + (size_t)(m1 * 16 + l15) * K;

    v8f a00 = {}, a01 = {}, a02 = {}, a03 = {};   // tile 0, subtiles 0..3
    v8f a10 = {}, a11 = {}, a12 = {}, a13 = {};   // tile 1, subtiles 0..3

    for (int k0 = 0; k0 < K; k0 += KCHUNK) {
        if (k0) __syncthreads();
        // ---- cooperative stage of B[k0:k0+128, nbase:nbase+64] in frag order
        for (int e = tid; e < (KCHUNK / 4) * 128; e += 256) {
            const int sub    = e & 3;
            const int lane_e = (e >> 2) & 31;
            const int kstep  = e >> 7;
            const int krow   = k0 + kstep * 4 + 2 * (lane_e >> 4);
            const int n      = nbase + sub * 16 + (lane_e & 15);
            float2 v;
            v.x = B[(size_t)krow * Nn + n];
            v.y = B[(size_t)(krow + 1) * Nn + n];
            ldsB[e] = v;
        }
        __syncthreads();

        // ---- compute over this K chunk (branch-free)
        for (int kstep = 0; kstep < KCHUNK / 4; ++kstep) {
            const int ka = k0 + kstep * 4 + 2 * half;
            const float4* bp = (const float4*)&ldsB[(kstep * 32 + lane) * 4];
            const float4 p0 = bp[0];
            const float4 p1 = bp[1];
            v2f f0, f1, f2, f3;
            f0.x = p0.x; f0.y = p0.y;
            f1.x = p0.z; f1.y = p0.w;
            f2.x = p1.x; f2.y = p1.y;
            f3.x = p1.z; f3.y = p1.w;

            const float2 t0 = *(const float2*)(Ap0 + ka);
            const float2 t1 = *(const float2*)(Ap1 + ka);
            v2f a0v; a0v.x = t0.x; a0v.y = t0.y;
            v2f a1v; a1v.x = t1.x; a1v.y = t1.y;

            a00 = WMMA_F32(a0v, f0, a00);
            a01 = WMMA_F32(a0v, f1, a01);
            a02 = WMMA_F32(a0v, f2, a02);
            a03 = WMMA_F32(a0v, f3, a03);
            a10 = WMMA_F32(a1v, f0, a10);
            a11 = WMMA_F32(a1v, f1, a11);
            a12 = WMMA_F32(a1v, f2, a12);
            a13 = WMMA_F32(a1v, f3, a13);
        }
    }

    // ---- store: VGPR i -> row i + 8*half, col = lane&15 (+16/32/48 per subtile)
    if (mt0 < mtiles) {
        float* Cp = C + (size_t)(mt0 * 16 + 8 * half) * Nn + nbase + l15;
        for (int i = 0; i < 8; ++i) {
            float* row = Cp + (size_t)i * Nn;
            if (accumulate) {
                row[0] += a00[i]; row[16] += a01[i]; row[32] += a02[i]; row[48] += a03[i];
            } else {
                row[0]  = a00[i]; row[16]  = a01[i]; row[32]  = a02[i]; row[48]  = a03[i];
            }
        }
    }
    if (mt1 < mtiles) {
        float* Cp = C + (size_t)(mt1 * 16 + 8 * half) * Nn + nbase + l15;
        for (int i = 0; i < 8; ++i) {
            float* row = Cp + (size_t)i * Nn;
            if (accumulate) {
                row[0] += a10[i]; row[16] += a11[i]; row[32] += a12[i]; row[48] += a13[i];
            } else {
                row[0]  = a10[i]; row[16]  = a11[i]; row[32]  = a12[i]; row[48]  = a13[i];
            }
        }
    }
}

// ---------------------------------------------------------------------------
// Per-node attention logits: al_s[i,h] = sum_c xp[i,h,c]*a_s[h,c] (wave/(i,h))
// ---------------------------------------------------------------------------
__global__ __launch_bounds__(256) void attn_logits_kernel(
    const float* __restrict__ xp, const float* __restrict__ a_s,
    const float* __restrict__ a_d, float* __restrict__ al_s,
    float* __restrict__ al_d, int n, int Hh, int Cc)
{
    const int wave = (int)((blockIdx.x * blockDim.x + threadIdx.x) >> 5);
    const int lane = threadIdx.x & 31;
    if (wave >= n * Hh) return;
    const int i = wave / Hh, h = wave % Hh;
    const float* xr = xp + (size_t)i * Hh * Cc + (size_t)h * Cc;
    const float* as = a_s + h * Cc;
    const float* ad = a_d + h * Cc;
    float ss = 0.f, sd = 0.f;
    for (int c = lane; c < Cc; c += 32) {
        const float v = xr[c];
        ss += v * as[c];
        sd += v * ad[c];
    }
    for (int off = 16; off; off >>= 1) {
        ss += __shfl_down(ss, off);
        sd += __shfl_down(sd, off);
    }
    if (lane == 0) { al_s[wave] = ss; al_d[wave] = sd; }
}

__device__ __forceinline__ float lrelu02(float x) { return x > 0.f ? x : 0.2f * x; }

// sign-aware float atomic max via integer atomics (monotone IEEE bit order)
__device__ __forceinline__ void atomicMaxFloat(float* addr, float val) {
    if (val >= 0.f) atomicMax((int*)addr, __float_as_int(val));
    else            atomicMin((unsigned int*)addr, __float_as_uint(val));
}

// m[i,h] = leaky_relu(al_s[i,h] + al_d[i,h])  (self loop seeds the segment max)
__global__ void self_max_init_kernel(const float* __restrict__ al_s,
                                     const float* __restrict__ al_d,
                                     float* __restrict__ mx, int n, int Hh)
{
    const int t = blockIdx.x * blockDim.x + threadIdx.x;
    if (t >= n * Hh) return;
    mx[t] = lrelu02(al_s[t] + al_d[t]);
}

__global__ void edge_max_kernel(const int* __restrict__ src, const int* __restrict__ dst,
                                const float* __restrict__ al_s, const float* __restrict__ al_d,
                                float* __restrict__ mx, int ne, int Hh)
{
    const int t = blockIdx.x * blockDim.x + threadIdx.x;
    if (t >= ne) return;
    const int s = src[t], d = dst[t];
    for (int h = 0; h < Hh; ++h) {
        const float e = lrelu02(al_s[s * Hh + h] + al_d[d * Hh + h]);
        atomicMaxFloat(&mx[d * Hh + h], e);
    }
}

// dn[i,h] = exp(e_self - m)   (self loop seeds the denominator)
__global__ void self_denom_init_kernel(const float* __restrict__ al_s,
                                       const float* __restrict__ al_d,
                                       const float* __restrict__ mx,
                                       float* __restrict__ dn, int n, int Hh)
{
    const int t = blockIdx.x * blockDim.x + threadIdx.x;
    if (t >= n * Hh) return;
    dn[t] = expf(lrelu02(al_s[t] + al_d[t]) - mx[t]);
}

__global__ void edge_denom_kernel(const int* __restrict__ src, const int* __restrict__ dst,
                                  const float* __restrict__ al_s, const float* __restrict__ al_d,
                                  const float* __restrict__ mx, float* __restrict__ dn,
                                  int ne, int Hh)
{
    const int t = blockIdx.x * blockDim.x + threadIdx.x;
    if (t >= ne) return;
    const int s = src[t], d = dst[t];
    for (int h = 0; h < Hh; ++h) {
        const float e = lrelu02(al_s[s * Hh + h] + al_d[d * Hh + h]);
        atomicAdd(&dn[d * Hh + h], expf(e - mx[d * Hh + h]));
    }
}

// agg[i,f] = alpha_self(i, f/C) * xp[i,f]  (self loop contribution)
__global__ void self_agg_init_kernel(const float* __restrict__ xp,
                                     const float* __restrict__ al_s,
                                     const float* __restrict__ al_d,
                                     const float* __restrict__ mx,
                                     const float* __restrict__ dn,
                                     float* __restrict__ agg, int n, int Hh, int Cc)
{
    const int t = blockIdx.x * blockDim.x + threadIdx.x;
    const int total = n * Hh * Cc;
    if (t >= total) return;
    const int i = t / (Hh * Cc);
    const int h = (t / Cc) % Hh;
    const int ih = i * Hh + h;
    const float e = lrelu02(al_s[ih] + al_d[ih]);
    const float alpha = expf(e - mx[ih]) / (dn[ih] + 1e-16f);
    agg[t] = alpha * xp[t];
}

// wave per edge: agg[dst] += alpha[h] * xp[src]   (float4 gathers, L2 atomics)
__global__ __launch_bounds__(256) void edge_agg_kernel(
    const int* __restrict__ src, const int* __restrict__ dst,
    const float* __restrict__ xp, const float* __restrict__ al_s,
    const float* __restrict__ al_d, const float* __restrict__ mx,
    const float* __restrict__ dn, float* __restrict__ agg,
    int ne, int Hh, int Cc)
{
    const int wave = (int)((blockIdx.x * blockDim.x + threadIdx.x) >> 5);
    const int lane = threadIdx.x & 31;
    if (wave >= ne) return;
    const int s = src[wave], d = dst[wave];
    float alpha[4];
    for (int h = 0; h < Hh; ++h) {
        const float e = lrelu02(al_s[s * Hh + h] + al_d[d * Hh + h]);
        alpha[h] = expf(e - mx[d * Hh + h]) / (dn[d * Hh + h] + 1e-16f);
    }
    const int F = Hh * Cc;
    const float* xs = xp + (size_t)s * F;
    float* ag = agg + (size_t)d * F;
    for (int base = lane * 4; base < F; base += 128) {
        const float4 v = *(const float4*)(xs + base);
        const float a = alpha[base / Cc];     // Cc % 4 == 0 -> same head
        atomicAdd(ag + base + 0, a * v.x);
        atomicAdd(ag + base + 1, a * v.y);
        atomicAdd(ag + base + 2, a * v.z);
        atomicAdd(ag + base + 3, a * v.w);
    }
}

// hacc += (elu?)(agg + bias)
__global__ void accum_kernel(const float* __restrict__ agg, const float* __restrict__ bias,
                             float* __restrict__ hacc, int n, int F, int do_elu)
{
    const int t = blockIdx.x * blockDim.x + threadIdx.x;
    if (t >= n * F) return;
    float v = agg[t] + bias[t % F];
    if (do_elu) v = v > 0.f ? v : (expf(v) - 1.f);
    hacc[t] += v;
}

// wave per row: LayerNorm over F=256 features; y already holds h2 + x0@res_W
__global__ __launch_bounds__(256) void ln_kernel(
    float* __restrict__ y, const float* __restrict__ res_b,
    const float* __restrict__ g, const float* __restrict__ bb,
    float* __restrict__ out, int n, int F)
{
    const int wave = (int)((blockIdx.x * blockDim.x + threadIdx.x) >> 5);
    const int lane = threadIdx.x & 31;
    if (wave >= n) return;
    float* row = y + (size_t)wave * F;
    const int per = F >> 5;                  // 8
    float vals[8];
    float s = 0.f, s2 = 0.f;
    for (int j = 0; j < per; ++j) {
        const int c = lane + 32 * j;
        const float v = row[c] + res_b[c];
        vals[j] = v; s += v; s2 += v * v;
    }
    for (int off = 16; off; off >>= 1) {
        s  += __shfl_down(s, off);
        s2 += __shfl_down(s2, off);
    }
    s  = __shfl(s, 0);
    s2 = __shfl(s2, 0);
    const float mu  = s / (float)F;
    const float var = s2 / (float)F - mu * mu;
    const float inv = rsqrtf(var + 1e-5f);
    float* o = out + (size_t)wave * F;
    for (int j = 0; j < per; ++j) {
        const int c = lane + 32 * j;
        o[c] = (vals[j] - mu) * inv * g[c] + bb[c];
    }
}

// ---------------------------------------------------------------------------
// Host orchestration
// ---------------------------------------------------------------------------
static inline void launch_gemm(const float* A, const float* B, float* C,
                               int M, int Nn, int K, int accum, hipStream_t st)
{
    const int mtiles = M >> 4;
    dim3 grid(Nn >> 6, (mtiles + 15) / 16);
    wmma_gemm_f32_kernel<<<grid, 256, 0, st>>>(A, B, C, M, Nn, K, accum);
}

static void run_gat_layer(const float* x, const float* W, const float* a_s,
                          const float* a_d, const float* bias,
                          const int* src, const int* dst,
                          float* xp, float* agg, float* al_s, float* al_d,
                          float* mx, float* dn, float* hacc,
                          int Kin, int Hh, int Cc, int do_elu, hipStream_t st)
{
    const int F = Hh * Cc;
    launch_gemm(x, W, xp, NN, F, Kin, 0, st);

    const int nwavesL = NN * Hh;
    attn_logits_kernel<<<(nwavesL + 7) / 8, 256, 0, st>>>(xp, a_s, a_d, al_s, al_d, NN, Hh, Cc);

    const int nh = NN * Hh;
    self_max_init_kernel<<<(nh + 255) / 256, 256, 0, st>>>(al_s, al_d, mx, NN, Hh);
    edge_max_kernel<<<(EE + 255) / 256, 256, 0, st>>>(src, dst, al_s, al_d, mx, EE, Hh);
    self_denom_init_kernel<<<(nh + 255) / 256, 256, 0, st>>>(al_s, al_d, mx, dn, NN, Hh);
    edge_denom_kernel<<<(EE + 255) / 256, 256, 0, st>>>(src, dst, al_s, al_d, mx, dn, EE, Hh);

    const int nf = NN * F;
    self_agg_init_kernel<<<(nf + 255) / 256, 256, 0, st>>>(xp, al_s, al_d, mx, dn, agg, NN, Hh, Cc);
    edge_agg_kernel<<<(EE + 7) / 8, 256, 0, st>>>(src, dst, xp, al_s, al_d, mx, dn, agg, EE, Hh, Cc);

    accum_kernel<<<(nf + 255) / 256, 256, 0, st>>>(agg, bias, hacc, NN, F, do_elu);
}

extern "C" void kernel_launch(void* const* d_in, const int* in_sizes, int n_in,
                              void* d_out, int out_size, void* d_ws, size_t ws_size,
                              hipStream_t stream)
{
    (void)in_sizes; (void)n_in; (void)ws_size;

    const float* x0      = (const float*)d_in[0];
    const float* W1      = (const float*)d_in[1];
    const float* a_src1  = (const float*)d_in[2];
    const float* a_dst1  = (const float*)d_in[3];
    const float* b1      = (const float*)d_in[4];
    const float* W2      = (const float*)d_in[5];
    const float* a_src2  = (const float*)d_in[6];
    const float* a_dst2  = (const float*)d_in[7];
    const float* b2      = (const float*)d_in[8];
    const float* res_W   = (const float*)d_in[9];
    const float* res_b   = (const float*)d_in[10];
    const float* ln_g    = (const float*)d_in[11];
    const float* ln_b    = (const float*)d_in[12];
    const int*   eidx    = (const int*)d_in[13];

    // workspace carve-out
    char* p = (char*)d_ws;
    float* xp   = (float*)p; p += (size_t)NN * F1 * sizeof(float);   // 102.4 MB
    float* agg  = (float*)p; p += (size_t)NN * F1 * sizeof(float);   // 102.4 MB
    float* h1   = (float*)p; p += (size_t)NN * F1 * sizeof(float);   // 102.4 MB
    float* al_s = (float*)p; p += (size_t)NN * HH * sizeof(float);
    float* al_d = (float*)p; p += (size_t)NN * HH * sizeof(float);
    float* mx   = (float*)p; p += (size_t)NN * HH * sizeof(float);
    float* dn   = (float*)p; p += (size_t)NN * HH * sizeof(float);

    float* y = (float*)d_out;   // d_out doubles as h2 / y accumulator [N, 256]

    hipMemsetAsync(h1, 0, (size_t)NN * F1 * sizeof(float), stream);
    hipMemsetAsync(y,  0, (size_t)out_size * sizeof(float), stream);

    // ---- layer 1: 4 heads x 128, ELU, sum over relations ----
    for (int r = 0; r < RR; ++r) {
        const int* src = eidx + (size_t)r * 2 * EE;
        const int* dst = src + EE;
        run_gat_layer(x0,
                      W1 + (size_t)r * DD * F1,
                      a_src1 + (size_t)r * HH * HID_,
                      a_dst1 + (size_t)r * HH * HID_,
                      b1 + (size_t)r * F1,
                      src, dst, xp, agg, al_s, al_d, mx, dn, h1,
                      DD, HH, HID_, /*elu=*/1, stream);
    }

    // ---- layer 2: 1 head x 256, no ELU, sum over relations ----
    for (int r = 0; r < RR; ++r) {
        const int* src = eidx + (size_t)r * 2 * EE;
        const int* dst = src + EE;
        run_gat_layer(h1,
                      W2 + (size_t)r * F1 * OUTD,
                      a_src2 + (size_t)r * OUTD,
                      a_dst2 + (size_t)r * OUTD,
                      b2 + (size_t)r * OUTD,
                      src, dst, xp, agg, al_s, al_d, mx, dn, y,
                      F1, 1, OUTD, /*elu=*/0, stream);
    }

    // ---- residual projection: y += x0 @ res_W ----
    launch_gemm(x0, res_W, y, NN, OUTD, DD, /*accumulate=*/1, stream);

    // ---- LayerNorm (adds res_b, writes d_out in place) ----
    ln_kernel<<<(NN + 7) / 8, 256, 0, stream>>>(y, res_b, ln_g, ln_b, y, NN, OUTD);
}